// MambaBackbone_65266323030047
// MI455X (gfx1250) — compile-verified
//
#include <hip/hip_runtime.h>
#include <math.h>

// Problem constants (from reference)
#define BB      2
#define LL      2048
#define DD      768
#define NLAYERS 2
#define DI      1536          // 2*D
#define NSTATE  16
#define KCONV   4
#define DTR     48            // DT = (D+15)/16
#define RPROJ   80            // DT + 2*N
#define MROWS   (BB*LL)       // 4096

#define KC      16            // K-chunk staged in LDS (divides 768, 1536, 48)
#define SA      20            // padded LDS row stride (floats): conflict-free ds_read_b64

typedef __attribute__((ext_vector_type(2))) float v2f;
typedef __attribute__((ext_vector_type(8))) float v8f;

__device__ __forceinline__ float silu_f(float x)     { return x / (1.0f + __expf(-x)); }
__device__ __forceinline__ float softplus_f(float x) { return (x > 20.0f) ? x : log1pf(__expf(x)); }

// ---- async global->LDS copy (CDNA5), with safe fallback --------------------
#if defined(__has_builtin)
#if __has_builtin(__builtin_amdgcn_global_load_async_to_lds_b128)
#define USE_ASYNC_LDS 1
#endif
#endif

#ifdef USE_ASYNC_LDS
typedef int v4i_vs __attribute__((__vector_size__(4 * sizeof(int))));
typedef __attribute__((address_space(1))) v4i_vs* gas_v4i; // global (AS1)
typedef __attribute__((address_space(3))) v4i_vs* las_v4i; // LDS    (AS3)

__device__ __forceinline__ void async_cp16(const float* g, float* l) {
    __builtin_amdgcn_global_load_async_to_lds_b128(
        (gas_v4i)(v4i_vs*)(void*)(const_cast<float*>(g)),
        (las_v4i)(v4i_vs*)(void*)l,
        0, 0);
}
__device__ __forceinline__ void wait_async0() {
#if __has_builtin(__builtin_amdgcn_s_wait_asynccnt)
    __builtin_amdgcn_s_wait_asynccnt(0);
#else
    asm volatile("s_wait_asynccnt 0x0" ::: "memory");
#endif
}
#endif

// ---------------------------------------------------------------------------
// x = inputs_embeds * mask
// ---------------------------------------------------------------------------
__global__ void mask_embed_kernel(const float* __restrict__ emb,
                                  const int*   __restrict__ mask,
                                  float*       __restrict__ x)
{
    int i = blockIdx.x * blockDim.x + threadIdx.x;
    if (i < MROWS * DD) {
        int row = i / DD;
        x[i] = emb[i] * (float)mask[row];
    }
}

// ---------------------------------------------------------------------------
// RMS norm: one 256-thread block per row of D=768
// ---------------------------------------------------------------------------
__global__ void rmsnorm_kernel(const float* __restrict__ x,
                               const float* __restrict__ w,
                               float*       __restrict__ out)
{
    __shared__ float red[256];
    int row = blockIdx.x;
    const float* xr = x + (size_t)row * DD;
    float ss = 0.0f;
    for (int d = threadIdx.x; d < DD; d += 256) { float v = xr[d]; ss += v * v; }
    red[threadIdx.x] = ss;
    __syncthreads();
    for (int off = 128; off > 0; off >>= 1) {
        if (threadIdx.x < off) red[threadIdx.x] += red[threadIdx.x + off];
        __syncthreads();
    }
    float scale = rsqrtf(red[0] / (float)DD + 1e-5f);
    float* orow = out + (size_t)row * DD;
    for (int d = threadIdx.x; d < DD; d += 256) orow[d] = xr[d] * scale * w[d];
}

// ---------------------------------------------------------------------------
// GEMM: C[M,N] = X[M,K] * W[N,K]^T  (+ optional residual Cadd)
// WMMA f32 16x16x4, LDS double-buffered K-chunks (KC=16).
// Block = 128 threads (4 waves) -> 64x64 output tile; wave w owns rows w*16..+15.
// ---------------------------------------------------------------------------
__global__ void gemm_wmma_f32(const float* __restrict__ X,
                              const float* __restrict__ W,
                              float*       __restrict__ C,
                              const float* __restrict__ Cadd,
                              int M, int N, int Kdim, int lda, int ldc, int do_add)
{
    __shared__ float As[2][64 * SA + 4];
    __shared__ float Bs[2][64 * SA + 4];

    const int tid   = threadIdx.x;
    const int lane  = tid & 31;
    const int wave  = tid >> 5;
    const int mlane = lane & 15;
    const int hi    = lane >> 4;

    const int mblk   = blockIdx.y * 64;
    const int n_base = blockIdx.x * 64;

    // staging map: 64 rows x 4 float4 per chunk; thread -> (row, quad), 2 rows/thread
    const int qoff = (tid & 3) * 4;
    const int r0   = tid >> 2;        // 0..31
    const int r1   = r0 + 32;
    int nb0 = n_base + r0; if (nb0 >= N) nb0 = N - 1;   // clamp tail (N mult of 16)
    int nb1 = n_base + r1; if (nb1 >= N) nb1 = N - 1;
    const float* gA0 = X + (size_t)(mblk + r0) * lda + qoff;
    const float* gA1 = X + (size_t)(mblk + r1) * lda + qoff;
    const float* gB0 = W + (size_t)nb0 * Kdim + qoff;
    const float* gB1 = W + (size_t)nb1 * Kdim + qoff;

    v8f acc[4];
#pragma unroll
    for (int j = 0; j < 4; ++j)
#pragma unroll
        for (int r = 0; r < 8; ++r) acc[j][r] = 0.0f;

    const int nchunks = Kdim / KC;

#ifdef USE_ASYNC_LDS
    async_cp16(gA0, &As[0][r0 * SA + qoff]);
    async_cp16(gA1, &As[0][r1 * SA + qoff]);
    async_cp16(gB0, &Bs[0][r0 * SA + qoff]);
    async_cp16(gB1, &Bs[0][r1 * SA + qoff]);
#else
    float4 ra0 = *(const float4*)gA0;
    float4 ra1 = *(const float4*)gA1;
    float4 rb0 = *(const float4*)gB0;
    float4 rb1 = *(const float4*)gB1;
#endif

    for (int c = 0; c < nchunks; ++c) {
        const int buf = c & 1;

#ifdef USE_ASYNC_LDS
        wait_async0();              // chunk c resident (this wave's copies)
        __syncthreads();            // ... and everyone else's
        if (c + 1 < nchunks) {      // prefetch next chunk while computing
            const int kk = (c + 1) * KC;
            async_cp16(gA0 + kk, &As[buf ^ 1][r0 * SA + qoff]);
            async_cp16(gA1 + kk, &As[buf ^ 1][r1 * SA + qoff]);
            async_cp16(gB0 + kk, &Bs[buf ^ 1][r0 * SA + qoff]);
            async_cp16(gB1 + kk, &Bs[buf ^ 1][r1 * SA + qoff]);
        }
#else
        *(float4*)&As[buf][r0 * SA + qoff] = ra0;
        *(float4*)&As[buf][r1 * SA + qoff] = ra1;
        *(float4*)&Bs[buf][r0 * SA + qoff] = rb0;
        *(float4*)&Bs[buf][r1 * SA + qoff] = rb1;
        __syncthreads();
        if (c + 1 < nchunks) {      // issue next chunk's global loads now
            const int kk = (c + 1) * KC;
            ra0 = *(const float4*)(gA0 + kk);
            ra1 = *(const float4*)(gA1 + kk);
            rb0 = *(const float4*)(gB0 + kk);
            rb1 = *(const float4*)(gB1 + kk);
        }
#endif

        // 4 k-blocks of 4 per chunk: 16 WMMAs
        const float* Ab = &As[buf][(wave * 16 + mlane) * SA + 2 * hi];
#pragma unroll
        for (int kb = 0; kb < 4; ++kb) {
            v2f a = *(const v2f*)(Ab + kb * 4);
#pragma unroll
            for (int j = 0; j < 4; ++j) {
                v2f b = *(const v2f*)(&Bs[buf][(j * 16 + mlane) * SA + 2 * hi + kb * 4]);
                acc[j] = __builtin_amdgcn_wmma_f32_16x16x4_f32(
                    false, a, false, b, (short)0, acc[j], false, false);
            }
        }
        __syncthreads();            // all waves done reading buf before reuse
    }

#pragma unroll
    for (int j = 0; j < 4; ++j) {
        int nt = n_base + j * 16;
        if (nt < N) {               // uniform per wave
            int ncol = nt + mlane;
#pragma unroll
            for (int r = 0; r < 8; ++r) {
                size_t idx = (size_t)(mblk + wave * 16 + r + 8 * hi) * ldc + ncol;
                float v = acc[j][r];
                if (do_add) v += Cadd[idx];
                C[idx] = v;
            }
        }
    }
}

// ---------------------------------------------------------------------------
// Depthwise causal conv (K=4) + bias + SiLU:  proj[:, :DI] -> u
// ---------------------------------------------------------------------------
__global__ void conv_silu_kernel(const float* __restrict__ proj,
                                 const float* __restrict__ cw,
                                 const float* __restrict__ cb,
                                 float*       __restrict__ u)
{
    int i = blockIdx.x * blockDim.x + threadIdx.x;
    if (i >= MROWS * DI) return;
    int c   = i % DI;
    int row = i / DI;          // b*L + l
    int l   = row % LL;
    float acc = cb[c];
#pragma unroll
    for (int j = 0; j < KCONV; ++j) {
        int ll = l - (KCONV - 1) + j;
        if (ll >= 0)
            acc += cw[c * KCONV + j] * proj[(size_t)(row - (KCONV - 1) + j) * (2 * DI) + c];
    }
    u[i] = silu_f(acc);
}

// ---------------------------------------------------------------------------
// Selective scan, software-pipelined: loads for t+1 issue before compute of t.
// 16 lanes = 16 states of one channel; 16 channels / block; grid = B*(DI/16).
// ---------------------------------------------------------------------------
__global__ void scan_kernel(const float* __restrict__ ssm,    // [M, 80]  (dt_raw | B | C)
                            const float* __restrict__ dtlin,  // [M, DI]
                            const float* __restrict__ u,      // [M, DI]
                            const float* __restrict__ proj,   // [M, 2*DI] (gate at +DI)
                            const float* __restrict__ A_log,  // [DI, N]
                            const float* __restrict__ dpb,    // [DI]
                            const float* __restrict__ Dskip,  // [DI]
                            float*       __restrict__ y)      // [M, DI]
{
    const int grp = threadIdx.x >> 4;
    const int n   = threadIdx.x & 15;
    const int blocks_per_b = DI / 16;
    const int b = blockIdx.x / blocks_per_b;
    const int d = (blockIdx.x % blocks_per_b) * 16 + grp;

    const float A    = -__expf(A_log[d * NSTATE + n]);
    const float bias = dpb[d];
    const float dsk  = Dskip[d];

    float s = 0.0f;
    const size_t rowbase = (size_t)b * LL;

    // prime the pipeline with t = 0
    float dlin = dtlin[rowbase * DI + d];
    float ut   = u[rowbase * DI + d];
    float Bt   = ssm[rowbase * RPROJ + DTR + n];
    float Ct   = ssm[rowbase * RPROJ + DTR + NSTATE + n];
    float g    = proj[rowbase * (2 * DI) + DI + d];

    for (int t = 0; t < LL; ++t) {
        float dlin_n = 0.f, ut_n = 0.f, Bt_n = 0.f, Ct_n = 0.f, g_n = 0.f;
        if (t + 1 < LL) {                       // issue next-step loads early
            const size_t r2 = rowbase + t + 1;
            dlin_n = dtlin[r2 * DI + d];
            ut_n   = u[r2 * DI + d];
            Bt_n   = ssm[r2 * RPROJ + DTR + n];
            Ct_n   = ssm[r2 * RPROJ + DTR + NSTATE + n];
            g_n    = proj[r2 * (2 * DI) + DI + d];
        }

        const float dtv = softplus_f(dlin + bias);
        s = __expf(dtv * A) * s + dtv * Bt * ut;

        float p = s * Ct;
        p += __shfl_xor(p, 8, 16);
        p += __shfl_xor(p, 4, 16);
        p += __shfl_xor(p, 2, 16);
        p += __shfl_xor(p, 1, 16);

        if (n == 0) {
            const size_t row = rowbase + t;
            y[row * DI + d] = (p + ut * dsk) * silu_f(g);
        }

        dlin = dlin_n; ut = ut_n; Bt = Bt_n; Ct = Ct_n; g = g_n;
    }
}

// ---------------------------------------------------------------------------
extern "C" void kernel_launch(void* const* d_in, const int* in_sizes, int n_in,
                              void* d_out, int out_size, void* d_ws, size_t ws_size,
                              hipStream_t stream)
{
    (void)in_sizes; (void)n_in; (void)out_size; (void)ws_size;

    const float* emb  = (const float*)d_in[0];
    const int*   mask = (const int*)  d_in[1];
    const float* ipw  = (const float*)d_in[2];   // [NL, 2*DI, D]
    const float* cw   = (const float*)d_in[3];   // [NL, DI, K]
    const float* cb   = (const float*)d_in[4];   // [NL, DI]
    const float* xpw  = (const float*)d_in[5];   // [NL, 80, DI]
    const float* dpw  = (const float*)d_in[6];   // [NL, DI, 48]
    const float* dpb  = (const float*)d_in[7];   // [NL, DI]
    const float* Alog = (const float*)d_in[8];   // [NL, DI, N]
    const float* Dsk  = (const float*)d_in[9];   // [NL, DI]
    const float* opw  = (const float*)d_in[10];  // [NL, D, DI]
    const float* nw   = (const float*)d_in[11];  // [NL, D]
    const float* nfw  = (const float*)d_in[12];  // [D]
    float* out = (float*)d_out;

    char* ws = (char*)d_ws;
    size_t off = 0;
    auto alloc = [&](size_t elems) { float* p = (float*)(ws + off); off += elems * sizeof(float); return p; };

    float* x    = alloc((size_t)MROWS * DD);
    float* xn   = alloc((size_t)MROWS * DD);
    float* proj = alloc((size_t)MROWS * 2 * DI);
    float* ub   = alloc((size_t)MROWS * DI);
    float* ssm  = alloc((size_t)MROWS * RPROJ);
    float* dtl  = alloc((size_t)MROWS * DI);
    float* yb   = alloc((size_t)MROWS * DI);

    {
        int n = MROWS * DD;
        mask_embed_kernel<<<(n + 255) / 256, 256, 0, stream>>>(emb, mask, x);
    }

    for (int l = 0; l < NLAYERS; ++l) {
        rmsnorm_kernel<<<MROWS, 256, 0, stream>>>(x, nw + (size_t)l * DD, xn);

        // proj = xn @ ipw^T      [4096 x 3072], K=768
        gemm_wmma_f32<<<dim3((2 * DI) / 64, MROWS / 64), 128, 0, stream>>>(
            xn, ipw + (size_t)l * 2 * DI * DD, proj, nullptr,
            MROWS, 2 * DI, DD, DD, 2 * DI, 0);

        {
            int n = MROWS * DI;
            conv_silu_kernel<<<(n + 255) / 256, 256, 0, stream>>>(
                proj, cw + (size_t)l * DI * KCONV, cb + (size_t)l * DI, ub);
        }

        // ssm = u @ xpw^T        [4096 x 80], K=1536
        gemm_wmma_f32<<<dim3((RPROJ + 63) / 64, MROWS / 64), 128, 0, stream>>>(
            ub, xpw + (size_t)l * RPROJ * DI, ssm, nullptr,
            MROWS, RPROJ, DI, DI, RPROJ, 0);

        // dtl = ssm[:, :48] @ dpw^T   [4096 x 1536], K=48, lda=80
        gemm_wmma_f32<<<dim3(DI / 64, MROWS / 64), 128, 0, stream>>>(
            ssm, dpw + (size_t)l * DI * DTR, dtl, nullptr,
            MROWS, DI, DTR, RPROJ, DI, 0);

        scan_kernel<<<BB * (DI / 16), 256, 0, stream>>>(
            ssm, dtl, ub, proj,
            Alog + (size_t)l * DI * NSTATE, dpb + (size_t)l * DI, Dsk + (size_t)l * DI, yb);

        // x = yb @ opw^T + x     [4096 x 768], K=1536 (residual fused)
        gemm_wmma_f32<<<dim3(DD / 64, MROWS / 64), 128, 0, stream>>>(
            yb, opw + (size_t)l * DD * DI, x, x,
            MROWS, DD, DI, DI, DD, 1);
    }

    rmsnorm_kernel<<<MROWS, 256, 0, stream>>>(x, nfw, out);
}